// SVDLinear_77240691851703
// MI455X (gfx1250) — compile-verified
//
#include <hip/hip_runtime.h>

typedef __attribute__((ext_vector_type(2))) float v2f;
typedef __attribute__((ext_vector_type(4))) float v4f;
typedef __attribute__((ext_vector_type(8))) float v8f;
typedef int v4i_t __attribute__((vector_size(16)));   // matches builtin's param type

#define D_OUT  4096
#define D_IN   4096
#define RANK   410
#define RP     416      // rank padded to multiple of KC
#define TOKENS 1024

#define BM    64        // block tile M (tokens)
#define BN    128       // block tile N (d_out)
#define KC    32        // K chunk per LDS stage
#define PITCH 36        // LDS row pitch (dwords): 144B = 16B-aligned, conflict-free b64 frag reads
#define NCHUNK (D_IN / KC + RP / KC)   // 128 + 13 = 141 fused K chunks

#define PPITCH 20       // pitch for the small rank-proj kernel (KC=16 there)

// ---------------------------------------------------------------------------
// Async global->LDS copy (GLOBAL_LOAD_ASYNC_TO_LDS_B128, ASYNCcnt-tracked).
// Falls back to a sync VGPR round-trip if the builtin is unavailable.
// ---------------------------------------------------------------------------
#if __has_builtin(__builtin_amdgcn_global_load_async_to_lds_b128)
#define HAS_ASYNC_LDS 1
#endif

__device__ __forceinline__ void async_b128(const float* g, float* l)
{
#ifdef HAS_ASYNC_LDS
  __builtin_amdgcn_global_load_async_to_lds_b128(
      (__attribute__((address_space(1))) v4i_t*)g,
      (__attribute__((address_space(3))) v4i_t*)l,
      0, 0);
#else
  *(v4f*)l = *(const v4f*)g;
#endif
}

__device__ __forceinline__ void wait_async0()
{
#ifdef HAS_ASYNC_LDS
#if __has_builtin(__builtin_amdgcn_s_wait_asynccnt)
  __builtin_amdgcn_s_wait_asynccnt(0);
#else
  asm volatile("s_wait_asynccnt 0x0" ::: "memory");
#endif
#endif
}

// ---------------------------------------------------------------------------
// out = input @ weight^T + P @ Upad^T   (fused concatenated-K fp32 WMMA GEMM)
// Block: 128 threads = 4 waves; each wave computes a 32x64 tile (2x4 v8f accs).
// Double-buffered LDS; async loads of chunk i+1 overlap 64 WMMAs on chunk i.
// ---------------------------------------------------------------------------
__global__ __launch_bounds__(128) void svd_gemm_main(
    const float* __restrict__ input,   // [TOKENS][D_IN]
    const float* __restrict__ weight,  // [D_OUT][D_IN]
    const float* __restrict__ P,       // [TOKENS][RP]
    const float* __restrict__ Upad,    // [D_OUT][RP]
    float* __restrict__ out)           // [TOKENS][D_OUT]
{
  __shared__ float sA[2][BM * PITCH];
  __shared__ float sB[2][BN * PITCH];

  const int tid  = threadIdx.x;
  const int lane = tid & 31;
  const int wave = tid >> 5;
  const int wm   = (wave & 1) * 32;   // wave M offset within block (2 waves in M)
  const int wn   = (wave >> 1) * 64;  // wave N offset within block (2 waves in N)
  const int m0   = blockIdx.y * BM;
  const int n0   = blockIdx.x * BN;
  const int l16  = lane & 15;
  const int lhi  = lane >> 4;         // 0 or 1

  v8f acc[2][4] = {};

  // staging decomposition: 4 threads per row, 8 floats (2 x b128) each
  const int srow = tid >> 2;          // 0..31
  const int scol = (tid & 3) * 8;     // 0,8,16,24

  // chunk i -> source matrices and K offset (phase 0: weight path, phase 1: low rank)
  auto chunk_src = [&](int i, const float*& ga, const float*& gb, int& ld, int& k0) {
    if (i < D_IN / KC) { ga = input; gb = weight; ld = D_IN; k0 = i * KC; }
    else               { ga = P;     gb = Upad;   ld = RP;   k0 = (i - D_IN / KC) * KC; }
  };

  // issue all 12 async b128 copies for one stage
  auto stage = [&](int buf, const float* gA, const float* gB, int ld, int k0) {
#pragma unroll
    for (int r = 0; r < BM; r += 32) {
      const int row = r + srow;
      const float* src = gA + (size_t)(m0 + row) * ld + k0 + scol;
      async_b128(src,     &sA[buf][row * PITCH + scol]);
      async_b128(src + 4, &sA[buf][row * PITCH + scol + 4]);
    }
#pragma unroll
    for (int r = 0; r < BN; r += 32) {
      const int row = r + srow;
      const float* src = gB + (size_t)(n0 + row) * ld + k0 + scol;
      async_b128(src,     &sB[buf][row * PITCH + scol]);
      async_b128(src + 4, &sB[buf][row * PITCH + scol + 4]);
    }
  };

  // prologue: fill buffer 0
  {
    const float *ga, *gb; int ld, k0;
    chunk_src(0, ga, gb, ld, k0);
    stage(0, ga, gb, ld, k0);
  }
  wait_async0();
  __syncthreads();

  for (int i = 0; i < NCHUNK; ++i) {
    const int buf = i & 1;

    // kick off async loads for the next chunk into the other buffer
    if (i + 1 < NCHUNK) {
      const float *ga, *gb; int ld, k0;
      chunk_src(i + 1, ga, gb, ld, k0);
      stage(buf ^ 1, ga, gb, ld, k0);
    }

    // --- compute: 8 K-steps of 4, 8 WMMAs each (64 WMMAs per stage) ---
#pragma unroll
    for (int ks = 0; ks < KC; ks += 4) {
      v2f afrag[2], bfrag[4];
#pragma unroll
      for (int mi = 0; mi < 2; ++mi)
        afrag[mi] = *(const v2f*)&sA[buf][(wm + mi * 16 + l16) * PITCH + ks + 2 * lhi];
#pragma unroll
      for (int ni = 0; ni < 4; ++ni)
        bfrag[ni] = *(const v2f*)&sB[buf][(wn + ni * 16 + l16) * PITCH + ks + 2 * lhi];
#pragma unroll
      for (int mi = 0; mi < 2; ++mi)
#pragma unroll
        for (int ni = 0; ni < 4; ++ni)
          acc[mi][ni] = __builtin_amdgcn_wmma_f32_16x16x4_f32(
              false, afrag[mi], false, bfrag[ni],
              (short)0, acc[mi][ni], false, false);
    }

    // make next buffer's async writes visible to all waves; retire reads of buf
    wait_async0();
    __syncthreads();
  }

  // --- epilogue: C/D layout (VGPR v -> row v + 8*lhi, col l16) -> row-major out ---
#pragma unroll
  for (int mi = 0; mi < 2; ++mi)
#pragma unroll
    for (int ni = 0; ni < 4; ++ni)
#pragma unroll
      for (int v = 0; v < 8; ++v) {
        const int r = m0 + wm + mi * 16 + v + 8 * lhi;
        const int c = n0 + wn + ni * 16 + l16;
        out[(size_t)r * D_OUT + c] = acc[mi][ni][v];
      }
}

// ---------------------------------------------------------------------------
// P[t][r] = sigma[r] * sum_k input[t][k] * V[k][r], zero-padded to RP columns
// Block: 128 threads = 4 waves; block tile 64(M) x 16(N); K = D_IN
// ---------------------------------------------------------------------------
__global__ __launch_bounds__(128) void svd_rank_proj(
    const float* __restrict__ input,  // [TOKENS][D_IN]
    const float* __restrict__ V,      // [D_IN][RANK]
    const float* __restrict__ sigma,  // [RANK]
    float* __restrict__ P)            // [TOKENS][RP]
{
  __shared__ float sA[64 * PPITCH];
  __shared__ float sV[16 * PPITCH];  // [k][r] layout

  const int tid  = threadIdx.x;
  const int lane = tid & 31;
  const int wave = tid >> 5;
  const int m0   = blockIdx.x * 64;
  const int r0   = blockIdx.y * 16;
  const int l16  = lane & 15;
  const int lhi  = lane >> 4;

  v8f acc = {};

  for (int k0 = 0; k0 < D_IN; k0 += 16) {
    // stage input 64x16: 128 threads x 2 float4
    {
      const int row = tid >> 1;
      const int col = (tid & 1) * 8;
      const float* src = input + (size_t)(m0 + row) * D_IN + k0 + col;
      *(v4f*)&sA[row * PPITCH + col]     = *(const v4f*)(src);
      *(v4f*)&sA[row * PPITCH + col + 4] = *(const v4f*)(src + 4);
    }
    // stage V tile 16(k) x 16(r), zero-padding r >= RANK: 128 threads x 2 floats
    {
      const int idx = tid * 2;        // 0..254, even
      const int kk  = idx >> 4;
      const int rr  = idx & 15;
#pragma unroll
      for (int u = 0; u < 2; ++u) {
        const int r = r0 + rr + u;
        sV[kk * PPITCH + rr + u] = (r < RANK)
            ? V[(size_t)(k0 + kk) * RANK + r] : 0.f;
      }
    }
    __syncthreads();

#pragma unroll
    for (int ks = 0; ks < 16; ks += 4) {
      v2f a = *(const v2f*)&sA[(wave * 16 + l16) * PPITCH + ks + 2 * lhi];
      v2f b;
      b[0] = sV[(ks + 2 * lhi)     * PPITCH + l16];
      b[1] = sV[(ks + 2 * lhi + 1) * PPITCH + l16];
      acc = __builtin_amdgcn_wmma_f32_16x16x4_f32(
          false, a, false, b, (short)0, acc, false, false);
    }
    __syncthreads();
  }

  const int c = r0 + l16;
  const float s = (c < RANK) ? sigma[c] : 0.f;
#pragma unroll
  for (int v = 0; v < 8; ++v) {
    const int row = m0 + wave * 16 + v + 8 * lhi;
    P[(size_t)row * RP + c] = acc[v] * s;
  }
}

// ---------------------------------------------------------------------------
// Upad[o][r] = (r < RANK) ? U[o][r] : 0
// ---------------------------------------------------------------------------
__global__ void pad_U(const float* __restrict__ U, float* __restrict__ Upad)
{
  const int i = blockIdx.x * blockDim.x + threadIdx.x;
  if (i >= D_OUT * RP) return;
  const int o = i / RP;
  const int r = i - o * RP;
  Upad[i] = (r < RANK) ? U[(size_t)o * RANK + r] : 0.f;
}

// ---------------------------------------------------------------------------
extern "C" void kernel_launch(void* const* d_in, const int* in_sizes, int n_in,
                              void* d_out, int out_size, void* d_ws, size_t ws_size,
                              hipStream_t stream)
{
  const float* input  = (const float*)d_in[0];  // [1024][4096]
  const float* U      = (const float*)d_in[1];  // [4096][410]
  const float* sigma  = (const float*)d_in[2];  // [410]
  const float* V      = (const float*)d_in[3];  // [4096][410]
  const float* weight = (const float*)d_in[4];  // [4096][4096]
  float* out = (float*)d_out;                   // [1024][4096]

  float* P    = (float*)d_ws;                   // [1024][416]  (~1.7 MB)
  float* Upad = P + (size_t)TOKENS * RP;        // [4096][416]  (~6.8 MB)

  // 1) pad U into workspace
  {
    const int n = D_OUT * RP;
    pad_U<<<(n + 255) / 256, 256, 0, stream>>>(U, Upad);
  }
  // 2) P = (input @ V) * sigma  (padded)
  {
    dim3 grid(TOKENS / 64, RP / 16);
    svd_rank_proj<<<grid, 128, 0, stream>>>(input, V, sigma, P);
  }
  // 3) out = input @ weight^T + P @ Upad^T
  {
    dim3 grid(D_OUT / BN, TOKENS / BM);
    svd_gemm_main<<<grid, 128, 0, stream>>>(input, weight, P, Upad, out);
  }
}